// MHA_82145544504061
// MI455X (gfx1250) — compile-verified
//
#include <hip/hip_runtime.h>

// ---- problem constants (from reference) ----
constexpr int kB  = 512;
constexpr int kH  = 8;
constexpr int kD  = 31;
constexpr int kT  = 4;
constexpr int kP  = 50;
constexpr int kC  = kH * kD;        // 248
constexpr int kN  = kP * kT;        // 200
constexpr int kDT = kD * kT;        // 124
constexpr int kR  = 2 * kP - 1;     // 99
constexpr int kM1 = 3 * kD;         // 93 rows of W per head (q|k|v)
constexpr float kScale = 0.17960530202677491f;  // 31^-0.5

// ---- LDS layout (floats). Total 77392 floats = 309.6 KB <= 320 KB WGP LDS ----
constexpr int X_OFF  = 0;                         // X_b  [248][200]
constexpr int X_STR  = 200;
constexpr int Q_OFF  = X_OFF + kC * X_STR;        // q    [50][128]  (p, dt)
constexpr int Q_STR  = 128;
constexpr int K_OFF  = Q_OFF + kP * Q_STR;        // k    [124][52]  (dt, p) pre-transposed
constexpr int K_STR  = 52;
constexpr int V_OFF  = K_OFF + kDT * K_STR;       // v    [50][128]  (p, dt)
constexpr int V_STR  = 128;
constexpr int S_OFF  = V_OFF + kP * V_STR;        // scores/attn [50][56] (cols 50..55 stay 0)
constexpr int S_STR  = 56;
constexpr int RL_OFF = S_OFF + kP * S_STR;        // rel logits [50][100]
constexpr int RL_STR = 100;
constexpr int B_OFF  = RL_OFF + kP * RL_STR;      // bias [744] staged once
constexpr int SMEM_FLOATS = B_OFF + 3 * kC;       // 77392
static_assert(SMEM_FLOATS * 4 <= 320 * 1024, "LDS budget exceeded");

typedef float v2f __attribute__((ext_vector_type(2)));
typedef float v8f __attribute__((ext_vector_type(8)));

__device__ __forceinline__ int iclamp(int v, int hi) { return v < hi ? v : hi; }

__global__ __launch_bounds__(256)
void mha_fused_wmma_kernel(const float* __restrict__ x,     // [512][248][50][4]
                           const float* __restrict__ w,     // [744][248]
                           const float* __restrict__ bqkv,  // [744]
                           const float* __restrict__ rel,   // [99][124]
                           float* __restrict__ out)         // [512][248][50][4]
{
    extern __shared__ float smem[];
    const int b     = blockIdx.x;
    const int tid   = threadIdx.x;
    const int wave  = tid >> 5;
    const int lane  = tid & 31;
    const int lhalf = lane >> 4;    // 0: lanes 0-15, 1: lanes 16-31
    const int lmod  = lane & 15;

    // ---------------- Phase 0: X_b -> LDS via CDNA5 async-to-LDS engine (ASYNCcnt),
    // bias -> LDS, and zero the S padding region.
    {
        for (int i = tid; i < 3 * kC; i += 256) smem[B_OFF + i] = bqkv[i];
        // zero S once: phase-4 A fragment reads k=50,51 there and phases 2/3
        // only ever rewrite cols < 50, so the padding stays zero.
        for (int i = tid; i < kP * S_STR; i += 256) smem[S_OFF + i] = 0.f;

        // bulk 198.4 KB copy: per-lane B128 async loads, no VGPR staging.
        const unsigned lds0 = (unsigned)(uintptr_t)(smem + X_OFF);   // LDS byte offset
        const unsigned long long gbase =
            (unsigned long long)(x + (size_t)b * (kC * kN));
        for (int i = tid; i < (kC * kN) / 4; i += 256) {
            const unsigned ldsaddr = lds0 + (unsigned)i * 16u;
            const unsigned long long ga = gbase + (unsigned long long)i * 16ull;
            asm volatile("global_load_async_to_lds_b128 %0, %1, off"
                         :: "v"(ldsaddr), "v"(ga) : "memory");
        }
        asm volatile("s_wait_asynccnt 0" ::: "memory");
    }
    __syncthreads();

    for (int h = 0; h < kH; ++h) {
        // ---------- Phase 1: qkv_head = W_head[93x248] @ X_b[248x200] + bias
        // 6 m-tiles x 7 n-pair-tiles (each pair = two 16-wide N tiles sharing A), K=248
        for (int t = wave; t < 6 * 7; t += 8) {
            const int m0  = (t / 7) * 16;
            const int n0a = (t % 7) * 32;
            const int n0b = n0a + 16;
            const int m   = m0 + lmod;
            // clamped indices: out-of-range rows/cols compute garbage that is
            // masked at the store, so unconditional loads are safe.
            const int mc  = iclamp(m, kM1 - 1);
            const int s0  = mc / kD, d0 = mc - s0 * kD;
            const float* wrow = w + (size_t)(s0 * kC + h * kD + d0) * kC;
            const int na  = iclamp(n0a + lmod, kN - 1);
            const int nb  = iclamp(n0b + lmod, kN - 1);
            v8f acc0 = {}, acc1 = {};
            #pragma unroll 2
            for (int kk = 0; kk < 62; ++kk) {
                const int k = kk * 4 + lhalf * 2;
                const v2f a = *reinterpret_cast<const v2f*>(wrow + k);
                v2f b0, b1;
                b0.x = smem[X_OFF + k       * X_STR + na];
                b0.y = smem[X_OFF + (k + 1) * X_STR + na];
                b1.x = smem[X_OFF + k       * X_STR + nb];
                b1.y = smem[X_OFF + (k + 1) * X_STR + nb];
                acc0 = __builtin_amdgcn_wmma_f32_16x16x4_f32(false, a, false, b0,
                                                             (short)0, acc0, false, false);
                acc1 = __builtin_amdgcn_wmma_f32_16x16x4_f32(false, a, false, b1,
                                                             (short)0, acc1, false, false);
            }
            // scatter both sub-tiles into q/k/v LDS with bias add (masked stores)
            #pragma unroll
            for (int e = 0; e < 2; ++e) {
                const int n = (e ? n0b : n0a) + lmod;
                if (n < kN) {
                    const int p = n >> 2, tt = n & 3;
                    const v8f acc = e ? acc1 : acc0;
                    #pragma unroll
                    for (int r = 0; r < 8; ++r) {
                        const int mm = m0 + r + (lhalf << 3);
                        if (mm < kM1) {
                            const int s = mm / kD, d = mm - s * kD;
                            const float val = acc[r] + smem[B_OFF + s * kC + h * kD + d];
                            const int dt = d * kT + tt;
                            if (s == 0)      smem[Q_OFF + p  * Q_STR + dt] = val;
                            else if (s == 1) smem[K_OFF + dt * K_STR + p ] = val;
                            else             smem[V_OFF + p  * V_STR + dt] = val;
                        }
                    }
                }
            }
        }
        __syncthreads();

        // ---------- Phase 2: dots = q@k (4x4 tiles) and RL = q@rel^T (4x7 tiles), K=124
        for (int t = wave; t < 16 + 28; t += 8) {
            int m0, n0; bool isRel;
            if (t < 16) { m0 = (t >> 2) * 16;  n0 = (t & 3) * 16;      isRel = false; }
            else        { const int u = t - 16; m0 = (u / 7) * 16; n0 = (u % 7) * 16; isRel = true; }
            const int m  = m0 + lmod;
            const int n  = n0 + lmod;
            const int mc = iclamp(m, kP - 1);
            const int nc = isRel ? iclamp(n, kR - 1) : iclamp(n, kP - 1);
            const float* relrow = rel + (size_t)nc * kDT;
            v8f acc = {};
            for (int kk = 0; kk < 31; ++kk) {
                const int k = kk * 4 + lhalf * 2;
                const v2f a = *reinterpret_cast<const v2f*>(smem + Q_OFF + mc * Q_STR + k);
                v2f bb;
                if (isRel) {
                    bb = *reinterpret_cast<const v2f*>(relrow + k);
                } else {
                    bb.x = smem[K_OFF + k       * K_STR + nc];
                    bb.y = smem[K_OFF + (k + 1) * K_STR + nc];
                }
                acc = __builtin_amdgcn_wmma_f32_16x16x4_f32(false, a, false, bb,
                                                            (short)0, acc, false, false);
            }
            const bool nv = isRel ? (n < kR) : (n < kP);
            if (nv) {
                #pragma unroll
                for (int r = 0; r < 8; ++r) {
                    const int mm = m0 + r + (lhalf << 3);
                    if (mm < kP) {
                        if (isRel) smem[RL_OFF + mm * RL_STR + n] = acc[r];
                        else       smem[S_OFF  + mm * S_STR  + n] = acc[r];
                    }
                }
            }
        }
        __syncthreads();

        // ---------- Phase 3: softmax over q of (dots + RL[p, q-p+49]) * SCALE
        // rel_to_abs reindex folds to: bias[p][q] = RL[p][q - p + P - 1]
        for (int p = wave; p < kP; p += 8) {
            const int q0 = lane;            // always < 50
            const int q1 = lane + 32;
            float s0 = (smem[S_OFF + p * S_STR + q0] +
                        smem[RL_OFF + p * RL_STR + (q0 - p + kP - 1)]) * kScale;
            float s1 = -3.0e38f;
            if (q1 < kP)
                s1 = (smem[S_OFF + p * S_STR + q1] +
                      smem[RL_OFF + p * RL_STR + (q1 - p + kP - 1)]) * kScale;
            float mx = fmaxf(s0, s1);
            #pragma unroll
            for (int dlt = 16; dlt >= 1; dlt >>= 1) mx = fmaxf(mx, __shfl_xor(mx, dlt, 32));
            float e0 = __expf(s0 - mx);
            float e1 = (q1 < kP) ? __expf(s1 - mx) : 0.f;
            float sum = e0 + e1;
            #pragma unroll
            for (int dlt = 16; dlt >= 1; dlt >>= 1) sum += __shfl_xor(sum, dlt, 32);
            const float inv = 1.0f / sum;
            smem[S_OFF + p * S_STR + q0] = e0 * inv;           // cols >= 50 untouched (stay 0)
            if (q1 < kP) smem[S_OFF + p * S_STR + q1] = e1 * inv;
        }
        __syncthreads();

        // ---------- Phase 4: out_head = attn[50x50] @ v[50x124]  (K=50 -> 13 steps,
        // S cols 50..51 are zero so the K overhang contributes nothing)
        for (int t = wave; t < 4 * 8; t += 8) {
            const int m0 = (t >> 3) * 16;
            const int n0 = (t & 7) * 16;
            const int m  = m0 + lmod;
            const int n  = n0 + lmod;            // <= 127 < V_STR, no clamp needed
            const int mc = iclamp(m, kP - 1);
            v8f acc = {};
            for (int kk = 0; kk < 13; ++kk) {
                const int k = kk * 4 + lhalf * 2;
                const v2f a = *reinterpret_cast<const v2f*>(smem + S_OFF + mc * S_STR + k);
                v2f bb;
                bb.x = smem[V_OFF + k       * V_STR + n];   // rows 50,51 read garbage, A==0 there
                bb.y = smem[V_OFF + (k + 1) * V_STR + n];
                acc = __builtin_amdgcn_wmma_f32_16x16x4_f32(false, a, false, bb,
                                                            (short)0, acc, false, false);
            }
            if (n < kDT) {
                const int d = n >> 2, tt = n & 3;   // dt = d*4 + t
                float* orow = out + (((size_t)b * kC + h * kD + d) * kP) * kT + tt;
                #pragma unroll
                for (int r = 0; r < 8; ++r) {
                    const int mm = m0 + r + (lhalf << 3);   // mm = p
                    if (mm < kP) __builtin_nontemporal_store(acc[r], &orow[mm * kT]);
                }
            }
        }
        __syncthreads();   // protect q/k/v/S/RL reuse by next head
    }
}

extern "C" void kernel_launch(void* const* d_in, const int* in_sizes, int n_in,
                              void* d_out, int out_size, void* d_ws, size_t ws_size,
                              hipStream_t stream) {
    (void)in_sizes; (void)n_in; (void)out_size; (void)d_ws; (void)ws_size;
    const float* x   = (const float*)d_in[0];
    const float* wq  = (const float*)d_in[1];
    const float* bq  = (const float*)d_in[2];
    const float* rel = (const float*)d_in[3];
    float* out = (float*)d_out;

    const int smem_bytes = SMEM_FLOATS * (int)sizeof(float);
    (void)hipFuncSetAttribute(reinterpret_cast<const void*>(mha_fused_wmma_kernel),
                              hipFuncAttributeMaxDynamicSharedMemorySize, smem_bytes);
    mha_fused_wmma_kernel<<<kB, 256, smem_bytes, stream>>>(x, wq, bq, rel, out);
}